// AFGCN_4320737100491
// MI455X (gfx1250) — compile-verified
//
#include <hip/hip_runtime.h>
#include <math.h>

typedef float v2f __attribute__((ext_vector_type(2)));
typedef float v8f __attribute__((ext_vector_type(8)));

#define HF 128          // feature width H1 / F_IN
#define BGRAPH 64       // graphs
#define SORTN 1024      // bitonic sort width (>= max nodes per graph = 1000)
#define EPW 4           // edges per wave in scatter kernel

// ------------------------------------------------------------------ utility
__global__ void k_fill_f32(float* p, float v, int n) {
  int i = blockIdx.x * blockDim.x + threadIdx.x;
  if (i < n) p[i] = v;
}
__global__ void k_fill_i32(int* p, int v, int n) {
  int i = blockIdx.x * blockDim.x + threadIdx.x;
  if (i < n) p[i] = v;
}
__global__ void k_copy_i32(int* d, const int* s, int n) {
  int i = blockIdx.x * blockDim.x + threadIdx.x;
  if (i < n) d[i] = s[i];
}
// dst = src (first) or dst += src
__global__ void k_acc_add(float* dst, const float* src, int n, int first) {
  int i = blockIdx.x * blockDim.x + threadIdx.x;
  if (i < n) dst[i] = first ? src[i] : (dst[i] + src[i]);
}

// ---------------------------------------------------- WMMA f32 GEMM kernel
// C[M,N] = act(A[M,K] @ W[K,N] + bias)  (row-major, M%16==0, N%64==0, K%4==0,
// K<=256). Block = 256 threads = 8 waves; block computes up to 128 rows x 64
// cols. The 64-wide W slice is staged in LDS, pair-interleaved so each wave's
// B fragment is one ds_load_b64. One wave -> 16x64 tile via 4 accumulators of
// v_wmma_f32_16x16x4_f32 (A fragment reused 4x).
__global__ void k_gemm_wmma(const float* __restrict__ A,
                            const float* __restrict__ W,
                            const float* __restrict__ bias,
                            float* __restrict__ C,
                            int M, int K, int N, int relu) {
  __shared__ float sW[16384];        // (K/2) x (64 cols x 2) pair-interleaved
  int NT = N >> 6;                   // 64-wide column tiles
  int blockRow = blockIdx.x / NT;
  int col0 = (blockIdx.x % NT) << 6;
  int tid = threadIdx.x;

  // cooperative load of W[:, col0:col0+64] into LDS:
  // sW[(k>>1)*128 + c*2 + (k&1)] = W[k][col0+c]
  for (int idx = tid; idx < (K << 6); idx += 256) {
    int k = idx >> 6, c = idx & 63;
    sW[((k >> 1) << 7) + (c << 1) + (k & 1)] = W[(size_t)k * N + col0 + c];
  }
  __syncthreads();

  int wave = tid >> 5, lane = tid & 31;
  int row0 = (blockRow * 8 + wave) << 4;
  if (row0 >= M) return;             // safe: no barriers after this point

  int lhalf = lane & 15;
  int khi   = (lane >> 4) << 1;      // lanes 0-15 -> K+0,K+1 ; 16-31 -> K+2,K+3

  v8f acc[4] = {};
  const float* Arow = A + (size_t)(row0 + lhalf) * K;

  for (int k = 0; k < K; k += 4) {
    v2f a;
    a.x = Arow[k + khi];
    a.y = Arow[k + khi + 1];
    const float* bp = &sW[(((k + khi) >> 1) << 7) + (lhalf << 1)];
#pragma unroll
    for (int j = 0; j < 4; ++j) {
      v2f b = *(const v2f*)(bp + (j << 5));   // cols j*16+lhalf, rows khi/khi+1
      acc[j] = __builtin_amdgcn_wmma_f32_16x16x4_f32(
          false, a, false, b, (short)0, acc[j], false, false);
    }
  }

  int rbase = row0 + ((lane >> 4) << 3);      // +8 for upper lane half
#pragma unroll
  for (int j = 0; j < 4; ++j) {
    int c = col0 + (j << 4) + lhalf;
    float bv = bias[c];
#pragma unroll
    for (int vi = 0; vi < 8; ++vi) {
      float v = acc[j][vi] + bv;
      if (relu) v = fmaxf(v, 0.0f);
      C[(size_t)(rbase + vi) * N + c] = v;
    }
  }
}

// ------------------------------------------------------------- propagation
__global__ void k_deg_scatter(const int* __restrict__ dst,
                              const float* __restrict__ emask,
                              float* __restrict__ deg, int E) {
  int e = blockIdx.x * blockDim.x + threadIdx.x;
  if (e < E) atomicAdd(&deg[dst[e]], emask[e]);
}
__global__ void k_finish_dinv(float* deg, int n) {
  int i = blockIdx.x * blockDim.x + threadIdx.x;
  if (i < n) deg[i] = rsqrtf(deg[i] + 1.0f);
}
// hout = hin * (1 + dinv^2)   (self-loop + incremental term)
__global__ void k_scale_self(const float* __restrict__ hin,
                             float* __restrict__ hout,
                             const float* __restrict__ dinv, int nNodes) {
  int i = blockIdx.x * blockDim.x + threadIdx.x;
  if (i >= nNodes * HF) return;
  float di = dinv[i >> 7];
  hout[i] = hin[i] * (1.0f + di * di);
}
// EPW edges per wave32: hout[dst] += hin[src] * dinv[src]*dinv[dst]*emask.
// Source rows are prefetched (global_prefetch_b8) before the atomic loop.
__global__ void k_prop_edges(const float* __restrict__ hin,
                             float* __restrict__ hout,
                             const int* __restrict__ src,
                             const int* __restrict__ dst,
                             const float* __restrict__ emask,
                             const float* __restrict__ dinv, int E) {
  int wid  = (int)((blockIdx.x * blockDim.x + threadIdx.x) >> 5);
  int lane = threadIdx.x & 31;
  int e0 = wid * EPW;
  if (e0 >= E) return;
  int e1 = e0 + EPW;
  if (e1 > E) e1 = E;
  for (int e = e0; e < e1; ++e) {
    __builtin_prefetch(hin + (size_t)src[e] * HF + lane * 4, 0, 0);
  }
  for (int e = e0; e < e1; ++e) {
    int s = src[e], d = dst[e];
    float c = emask[e] * dinv[s] * dinv[d];
    if (c == 0.0f) continue;
    const float4* hp = (const float4*)(hin + (size_t)s * HF);
    float4 v = hp[lane];
    float* op = hout + (size_t)d * HF + lane * 4;
    atomicAdd(op + 0, v.x * c);
    atomicAdd(op + 1, v.y * c);
    atomicAdd(op + 2, v.z * c);
    atomicAdd(op + 3, v.w * c);
  }
}

// ------------------------------------------------------------ TopK pooling
__global__ void k_score(const float* __restrict__ x, const float* __restrict__ p,
                        float* __restrict__ score, int nNodes) {
  __shared__ float sp[HF];
  __shared__ float snorm;
  int t = threadIdx.x;
  if (t < HF) sp[t] = p[t];
  __syncthreads();
  if (t == 0) {
    float s = 0.0f;
    for (int j = 0; j < HF; ++j) s += sp[j] * sp[j];
    snorm = rsqrtf(s);
  }
  __syncthreads();
  int i = blockIdx.x * blockDim.x + t;
  if (i < nNodes) {
    const float* xr = x + (size_t)i * HF;
    float s = 0.0f;
    for (int j = 0; j < HF; ++j) s += xr[j] * sp[j];
    score[i] = s * snorm;
  }
}

// one block per graph: bitonic sort (desc, stable by index) of <=1024 scores
__global__ void k_topk(const float* __restrict__ score, int n, int k,
                       int* __restrict__ remap, int* __restrict__ perm,
                       float* __restrict__ gate) {
  __shared__ float skey[SORTN];
  __shared__ int   sidx[SORTN];
  int g = blockIdx.x;
  for (int i = threadIdx.x; i < SORTN; i += blockDim.x) {
    skey[i] = (i < n) ? score[g * n + i] : -__builtin_inff();
    sidx[i] = i;
  }
  __syncthreads();
  for (int sz = 2; sz <= SORTN; sz <<= 1) {
    for (int st = sz >> 1; st > 0; st >>= 1) {
      for (int i = threadIdx.x; i < SORTN; i += blockDim.x) {
        int j = i ^ st;
        if (j > i) {
          float a = skey[i], b = skey[j];
          int ia = sidx[i], ib = sidx[j];
          bool aFirst = (a > b) || (a == b && ia < ib);   // descending predicate
          bool desc = ((i & sz) == 0);
          if (desc ? !aFirst : aFirst) {
            skey[i] = b; skey[j] = a;
            sidx[i] = ib; sidx[j] = ia;
          }
        }
      }
      __syncthreads();
    }
  }
  for (int j = threadIdx.x; j < k; j += blockDim.x) {
    int orig = g * n + sidx[j];
    int nid  = g * k + j;
    remap[orig] = nid;
    perm[nid]   = orig;
    gate[nid]   = tanhf(skey[j]);
  }
}

__global__ void k_gather_gate(const float* __restrict__ hs,
                              const int* __restrict__ perm,
                              const float* __restrict__ gate,
                              float* __restrict__ xnew, int nNew) {
  int i = blockIdx.x * blockDim.x + threadIdx.x;
  if (i >= nNew * HF) return;
  int node = i >> 7, f = i & 127;
  xnew[i] = hs[(size_t)perm[node] * HF + f] * gate[node];
}

__global__ void k_remap_edges(int* src, int* dst, float* emask,
                              const int* __restrict__ remap, int E) {
  int e = blockIdx.x * blockDim.x + threadIdx.x;
  if (e >= E) return;
  int ns = remap[src[e]], nd = remap[dst[e]];
  float m = emask[e];
  if (ns < 0 || nd < 0) m = 0.0f;
  emask[e] = m;
  src[e] = ns < 0 ? 0 : ns;
  dst[e] = nd < 0 ? 0 : nd;
}

// per-graph max+mean over k kept nodes, accumulated into z[B,256]
__global__ void k_readout(const float* __restrict__ x, float* __restrict__ z, int k) {
  int g = blockIdx.x, f = threadIdx.x;   // 128 threads
  float mx = -__builtin_inff(), sm = 0.0f;
  for (int j = 0; j < k; ++j) {
    float v = x[(size_t)(g * k + j) * HF + f];
    mx = fmaxf(mx, v);
    sm += v;
  }
  z[g * 256 + f]       += mx;
  z[g * 256 + 128 + f] += sm / (float)k;
}

// final: out[64,10] = log_softmax(t2[64,64] @ w3[64,10] + b3)
__global__ void k_mlp_final(const float* __restrict__ t2,
                            const float* __restrict__ w3,
                            const float* __restrict__ b3,
                            float* __restrict__ out) {
  int r = threadIdx.x;   // 64 rows
  float l[10];
  for (int c = 0; c < 10; ++c) {
    float s = b3[c];
    for (int j = 0; j < 64; ++j) s += t2[r * 64 + j] * w3[j * 10 + c];
    l[c] = s;
  }
  float m = l[0];
  for (int c = 1; c < 10; ++c) m = fmaxf(m, l[c]);
  float se = 0.0f;
  for (int c = 0; c < 10; ++c) se += expf(l[c] - m);
  float lse = logf(se);
  for (int c = 0; c < 10; ++c) out[r * 10 + c] = l[c] - m - lse;
}

// ------------------------------------------------------------------ driver
static inline int cdiv(long long a, int b) { return (int)((a + b - 1) / b); }

extern "C" void kernel_launch(void* const* d_in, const int* in_sizes, int n_in,
                              void* d_out, int out_size, void* d_ws, size_t ws_size,
                              hipStream_t stream) {
  const float* x0   = (const float*)d_in[0];
  const int*   src0 = (const int*)d_in[1];
  const int*   dst0 = (const int*)d_in[2];
  const int E       = in_sizes[1];
  const int nodes0  = in_sizes[0] / HF;           // 64000
  const int nPer0   = nodes0 / BGRAPH;            // 1000

  // weight/bias input indices: layer L branch b -> w at widx, bias at widx+1
  const int widx[3][3] = {{3, 5, 7}, {9, 11, 13}, {15, 17, 19}};
  const int pidx[3] = {27, 28, 29};

  // ---- carve workspace
  char* wp = (char*)d_ws;
  auto carve = [&](size_t bytes) {
    void* p = (void*)wp;
    wp += (bytes + 255) & ~(size_t)255;
    return p;
  };
  size_t nodeBuf = (size_t)nodes0 * HF * sizeof(float);
  float* bufA  = (float*)carve(nodeBuf);
  float* bufB  = (float*)carve(nodeBuf);
  float* bufS  = (float*)carve(nodeBuf);          // branch sum h
  float* bufX  = (float*)carve(nodeBuf);          // pooled features
  int*   srcE  = (int*)carve((size_t)E * 4);
  int*   dstE  = (int*)carve((size_t)E * 4);
  float* emask = (float*)carve((size_t)E * 4);
  float* dinv  = (float*)carve((size_t)nodes0 * 4);
  float* score = (float*)carve((size_t)nodes0 * 4);
  float* gate  = (float*)carve((size_t)nodes0 * 4);
  int*   remap = (int*)carve((size_t)nodes0 * 4);
  int*   perm  = (int*)carve((size_t)nodes0 * 4);
  float* z     = (float*)carve((size_t)BGRAPH * 256 * 4);
  float* t1    = (float*)carve((size_t)BGRAPH * 128 * 4);
  float* t2    = (float*)carve((size_t)BGRAPH * 64 * 4);

  const int TB = 256;

  // ---- init mutable edge state + readout accumulator
  k_copy_i32<<<cdiv(E, TB), TB, 0, stream>>>(srcE, src0, E);
  k_copy_i32<<<cdiv(E, TB), TB, 0, stream>>>(dstE, dst0, E);
  k_fill_f32<<<cdiv(E, TB), TB, 0, stream>>>(emask, 1.0f, E);
  k_fill_f32<<<cdiv(BGRAPH * 256, TB), TB, 0, stream>>>(z, 0.0f, BGRAPH * 256);

  const float* xin = x0;
  int nPer = nPer0;

  for (int L = 0; L < 3; ++L) {
    int nNodes = BGRAPH * nPer;
    int kPer = (int)((4 * nPer + 4) / 5);         // ceil(0.8 * nPer)
    int nNew = BGRAPH * kPer;

    // degree -> dinv (shared across branches/steps of this layer)
    k_fill_f32<<<cdiv(nNodes, TB), TB, 0, stream>>>(dinv, 0.0f, nNodes);
    k_deg_scatter<<<cdiv(E, TB), TB, 0, stream>>>(dstE, emask, dinv, E);
    k_finish_dinv<<<cdiv(nNodes, TB), TB, 0, stream>>>(dinv, nNodes);

    long long featN = (long long)nNodes * HF;
    int gFeat  = cdiv(featN, TB);
    int gEdgeW = cdiv((long long)cdiv(E, EPW) * 32, TB);  // EPW edges per wave32

    for (int b = 0; b < 3; ++b) {
      const float* W    = (const float*)d_in[widx[L][b]];
      const float* bias = (const float*)d_in[widx[L][b] + 1];
      // h0 = relu(xin @ W + b)   — block covers 128 rows x 64 cols
      int gGemm = cdiv(nNodes, 128) * (HF / 64);
      k_gemm_wmma<<<gGemm, TB, 0, stream>>>(xin, W, bias, bufA, nNodes, HF, HF, 1);
      // gra_inc: 2 incremental propagation steps
      k_scale_self<<<gFeat, TB, 0, stream>>>(bufA, bufB, dinv, nNodes);
      k_prop_edges<<<gEdgeW, TB, 0, stream>>>(bufA, bufB, srcE, dstE, emask, dinv, E);
      k_scale_self<<<gFeat, TB, 0, stream>>>(bufB, bufA, dinv, nNodes);
      k_prop_edges<<<gEdgeW, TB, 0, stream>>>(bufB, bufA, srcE, dstE, emask, dinv, E);
      // accumulate branch
      k_acc_add<<<gFeat, TB, 0, stream>>>(bufS, bufA, (int)featN, b == 0);
    }

    // ---- TopK pooling on bufS
    k_score<<<cdiv(nNodes, TB), TB, 0, stream>>>(bufS, (const float*)d_in[pidx[L]],
                                                 score, nNodes);
    k_fill_i32<<<cdiv(nNodes, TB), TB, 0, stream>>>(remap, -1, nNodes);
    k_topk<<<BGRAPH, 512, 0, stream>>>(score, nPer, kPer, remap, perm, gate);
    k_gather_gate<<<cdiv((long long)nNew * HF, TB), TB, 0, stream>>>(
        bufS, perm, gate, bufX, nNew);
    k_remap_edges<<<cdiv(E, TB), TB, 0, stream>>>(srcE, dstE, emask, remap, E);
    k_readout<<<BGRAPH, 128, 0, stream>>>(bufX, z, kPer);

    xin = bufX;
    nPer = kPer;
  }

  // ---- final MLP (WMMA GEMMs) + log_softmax
  {
    const float* w1 = (const float*)d_in[21];
    const float* b1 = (const float*)d_in[22];
    const float* w2 = (const float*)d_in[23];
    const float* b2 = (const float*)d_in[24];
    const float* w3 = (const float*)d_in[25];
    const float* b3 = (const float*)d_in[26];
    k_gemm_wmma<<<cdiv(BGRAPH, 128) * (128 / 64), TB, 0, stream>>>(
        z, w1, b1, t1, BGRAPH, 256, 128, 1);
    k_gemm_wmma<<<cdiv(BGRAPH, 128) * (64 / 64), TB, 0, stream>>>(
        t1, w2, b2, t2, BGRAPH, 128, 64, 1);
    k_mlp_final<<<1, 64, 0, stream>>>(t2, w3, b3, (float*)d_out);
  }
}